// CPUMamba_71743133712898
// MI455X (gfx1250) — compile-verified
//
#include <hip/hip_runtime.h>
#include <hip/hip_bf16.h>
#include <math.h>

// ---------------------------------------------------------------------------
// Mamba block for MI455X (gfx1250, wave32).
// GEMMs via V_WMMA_F32_16X16X4_F32; scan parallelized one lane per (b,d,s)
// recurrence (65536 lanes) with 16-lane shuffle reductions.
// ---------------------------------------------------------------------------

typedef float v2f __attribute__((ext_vector_type(2)));
typedef float v8f __attribute__((ext_vector_type(8)));

constexpr int DM   = 1024;   // d_model
constexpr int DS   = 16;     // d_state
constexpr int DCV  = 4;      // d_conv
constexpr int DI   = 2048;   // d_inner
constexpr int DTR  = 64;     // dt_rank
constexpr int BSZ  = 2;
constexpr int TLEN = 2048;
constexpr int MTOT = BSZ * TLEN;        // 4096 rows for all GEMMs
constexpr int XDBL_W = DTR + 2 * DS;    // 96

__device__ __forceinline__ float dev_silu(float v) {
    return v / (1.0f + __expf(-v));
}
__device__ __forceinline__ float dev_softplus(float v) {
    return (v > 20.0f) ? v : log1pf(__expf(v));
}

// ---------------------------------------------------------------------------
// NT-GEMM (no bounds checks): C[M,N] = A[M,K] * W[N,K]^T, both row-major.
// Wave tile: 32(M) x 64(N) -> 8 f32 WMMA accumulators, 6 b64 loads / 8 WMMAs.
// Block: 256 threads = 8 waves arranged 2(M) x 4(N) -> 64 x 256 block tile.
// Requires: M % 64 == 0, N % 256 == 0 (true at all call sites).
// EPI: 0 = plain store, 1 = softplus(acc + bias[n])  (dt projection)
// ---------------------------------------------------------------------------
template <int EPI>
__global__ __launch_bounds__(256) void gemm_nt_wmma(
    const float* __restrict__ A, int lda,
    const float* __restrict__ W, int ldb,
    float* __restrict__ C, int ldc,
    int K,
    const float* __restrict__ bias)
{
    const int lane = threadIdx.x & 31;
    const int wave = threadIdx.x >> 5;
    const int lm   = lane & 15;       // M index (A), N index (B/C)
    const int half = lane >> 4;       // selects K pair / M+8 rows of C
    const int koff = half * 2;

    const int m0 = blockIdx.y * 64 + (wave >> 2) * 32;
    const int n0 = blockIdx.x * 256 + (wave & 3) * 64;

    const float* arow0 = A + (size_t)(m0 + lm) * lda + koff;
    const float* arow1 = arow0 + (size_t)16 * lda;
    const float* brow0 = W + (size_t)(n0 + lm) * ldb + koff;
    const float* brow1 = brow0 + (size_t)16 * ldb;
    const float* brow2 = brow0 + (size_t)32 * ldb;
    const float* brow3 = brow0 + (size_t)48 * ldb;

    v8f acc[2][4] = {};

#pragma unroll 2
    for (int k = 0; k < K; k += 4) {
        v2f a0 = *(const v2f*)(arow0 + k);
        v2f a1 = *(const v2f*)(arow1 + k);
        v2f b0 = *(const v2f*)(brow0 + k);
        v2f b1 = *(const v2f*)(brow1 + k);
        v2f b2 = *(const v2f*)(brow2 + k);
        v2f b3 = *(const v2f*)(brow3 + k);
        acc[0][0] = __builtin_amdgcn_wmma_f32_16x16x4_f32(false, a0, false, b0, (short)0, acc[0][0], false, false);
        acc[0][1] = __builtin_amdgcn_wmma_f32_16x16x4_f32(false, a0, false, b1, (short)0, acc[0][1], false, false);
        acc[0][2] = __builtin_amdgcn_wmma_f32_16x16x4_f32(false, a0, false, b2, (short)0, acc[0][2], false, false);
        acc[0][3] = __builtin_amdgcn_wmma_f32_16x16x4_f32(false, a0, false, b3, (short)0, acc[0][3], false, false);
        acc[1][0] = __builtin_amdgcn_wmma_f32_16x16x4_f32(false, a1, false, b0, (short)0, acc[1][0], false, false);
        acc[1][1] = __builtin_amdgcn_wmma_f32_16x16x4_f32(false, a1, false, b1, (short)0, acc[1][1], false, false);
        acc[1][2] = __builtin_amdgcn_wmma_f32_16x16x4_f32(false, a1, false, b2, (short)0, acc[1][2], false, false);
        acc[1][3] = __builtin_amdgcn_wmma_f32_16x16x4_f32(false, a1, false, b3, (short)0, acc[1][3], false, false);
    }

    // C/D layout: VGPR v -> row (tile_m + v + 8*half) ; lane%16 -> col
#pragma unroll
    for (int j = 0; j < 4; ++j) {
        int col = n0 + j * 16 + lm;
        float bv = (EPI == 1) ? bias[col] : 0.0f;
#pragma unroll
        for (int i = 0; i < 2; ++i) {
#pragma unroll
            for (int v = 0; v < 8; ++v) {
                int row = m0 + i * 16 + v + half * 8;
                float val = acc[i][j][v];
                if (EPI == 1) val = dev_softplus(val + bv);
                C[(size_t)row * ldc + col] = val;
            }
        }
    }
}

// ---------------------------------------------------------------------------
// Ragged-N NT-GEMM (branchless guards, clamped loads + v_cndmask zeroing).
// Wave tile 16x64; used only for x_dbl (N=96). M % 32 == 0 required.
// ---------------------------------------------------------------------------
__global__ __launch_bounds__(256) void gemm_nt_wmma_ragged(
    const float* __restrict__ A, int lda,
    const float* __restrict__ W, int ldb,
    float* __restrict__ C, int ldc,
    int N, int K)
{
    const int lane = threadIdx.x & 31;
    const int wave = threadIdx.x >> 5;
    const int lm   = lane & 15;
    const int half = lane >> 4;
    const int koff = half * 2;

    const int m0 = blockIdx.y * 32 + (wave >> 2) * 16;
    const int n0 = blockIdx.x * 256 + (wave & 3) * 64;
    if (n0 >= N) return;

    const float* arow = A + (size_t)(m0 + lm) * lda + koff;

    const float* brow[4];
    bool bok[4];
#pragma unroll
    for (int j = 0; j < 4; ++j) {
        int n  = n0 + j * 16 + lm;
        bok[j] = (n < N);
        brow[j] = W + (size_t)(bok[j] ? n : 0) * ldb + koff;  // clamp: in-bounds
    }

    v8f acc[4] = {};

#pragma unroll 2
    for (int k = 0; k < K; k += 4) {
        v2f a = *(const v2f*)(arow + k);
        v2f b[4];
#pragma unroll
        for (int j = 0; j < 4; ++j) {
            v2f t = *(const v2f*)(brow[j] + k);     // always safe (clamped)
            b[j].x = bok[j] ? t.x : 0.0f;           // v_cndmask, no EXEC branch
            b[j].y = bok[j] ? t.y : 0.0f;
        }
#pragma unroll
        for (int j = 0; j < 4; ++j) {
            acc[j] = __builtin_amdgcn_wmma_f32_16x16x4_f32(
                false, a, false, b[j], (short)0, acc[j], false, false);
        }
    }

#pragma unroll
    for (int j = 0; j < 4; ++j) {
        int col = n0 + j * 16 + lm;
        if (col >= N) continue;
#pragma unroll
        for (int v = 0; v < 8; ++v) {
            int row = m0 + v + half * 8;
            C[(size_t)row * ldc + col] = acc[j][v];
        }
    }
}

// ---------------------------------------------------------------------------
// Depthwise causal conv1d (width 4) + bias + SiLU over the x half of xz.
// xz: (B,T,2*DI) row-major; x_act: (B,T,DI)
// ---------------------------------------------------------------------------
__global__ __launch_bounds__(256) void conv_silu_kernel(
    const float* __restrict__ xz,
    const float* __restrict__ conv_w,  // (DI,1,4)
    const float* __restrict__ conv_b,  // (DI)
    float* __restrict__ x_act)
{
    int idx = blockIdx.x * 256 + threadIdx.x;     // B*T*DI total
    int d = idx & (DI - 1);
    int r = idx >> 11;                            // DI == 2048
    int t = r & (TLEN - 1);                       // TLEN == 2048
    int b = r >> 11;

    const float* w = conv_w + d * DCV;
    float acc = conv_b[d];
#pragma unroll
    for (int j = 0; j < DCV; ++j) {
        int tj = t - (DCV - 1) + j;
        if (tj >= 0) {
            float xv = xz[((size_t)(b * TLEN + tj)) * (2 * DI) + d];
            acc = fmaf(w[j], xv, acc);
        }
    }
    x_act[(size_t)idx] = dev_silu(acc);
}

// ---------------------------------------------------------------------------
// Selective scan: one lane per (b, d, s) recurrence; 16-lane groups share d.
// h_t = exp(dt*A)*h + dt*B_t*x ; y = sum_s h*C_t ; y = (y + D*x)*silu(z)
// yio holds x_act on entry; gated y is written in place (same-wave RAW safe).
// ---------------------------------------------------------------------------
__global__ __launch_bounds__(256) void scan_kernel(
    const float* __restrict__ dtb,    // (B,T,DI) softplus'ed dt
    const float* __restrict__ xdbl,   // (B,T,96): [dt_r | B | C]
    const float* __restrict__ xz,     // (B,T,2*DI), z half read here
    const float* __restrict__ A_log,  // (DI,DS)
    const float* __restrict__ Dvec,   // (DI)
    float* __restrict__ yio)          // in: x_act ; out: gated y
{
    int g = blockIdx.x * 256 + threadIdx.x;       // B*DI*DS lanes total
    int s = g & (DS - 1);
    int d = (g >> 4) & (DI - 1);
    int b = g >> 15;                              // / (DS*DI)

    const float Aval = -__expf(A_log[d * DS + s]);
    const float Dv   = Dvec[d];

    const float* dtp = dtb  + (size_t)b * TLEN * DI + d;
    const float* bp  = xdbl + (size_t)b * TLEN * XDBL_W + DTR + s;
    const float* cp  = xdbl + (size_t)b * TLEN * XDBL_W + DTR + DS + s;
    const float* zp  = xz   + (size_t)b * TLEN * (2 * DI) + DI + d;
    float*       yp  = yio  + (size_t)b * TLEN * DI + d;

    float h = 0.0f;
    for (int t = 0; t < TLEN; ++t) {
        float dt = dtp[(size_t)t * DI];
        float xv = yp [(size_t)t * DI];           // x_act (pre-overwrite)
        float Bv = bp [(size_t)t * XDBL_W];
        float Cv = cp [(size_t)t * XDBL_W];

        float dA = __expf(dt * Aval);
        h = fmaf(h, dA, dt * Bv * xv);
        float p = h * Cv;

        // reduce over the 16 states held by this 16-lane group
        p += __shfl_xor(p, 1, 32);
        p += __shfl_xor(p, 2, 32);
        p += __shfl_xor(p, 4, 32);
        p += __shfl_xor(p, 8, 32);

        if (s == 0) {
            float zv = zp[(size_t)t * (2 * DI)];
            float y  = p + Dv * xv;
            yp[(size_t)t * DI] = y * dev_silu(zv);
        }
    }
}

// ---------------------------------------------------------------------------
extern "C" void kernel_launch(void* const* d_in, const int* in_sizes, int n_in,
                              void* d_out, int out_size, void* d_ws, size_t ws_size,
                              hipStream_t stream) {
    (void)in_sizes; (void)n_in; (void)out_size; (void)ws_size;

    const float* hs     = (const float*)d_in[0];  // (B,T,DM)
    const float* W_in   = (const float*)d_in[1];  // (2*DI, DM)
    const float* conv_w = (const float*)d_in[2];  // (DI,1,4)
    const float* conv_b = (const float*)d_in[3];  // (DI)
    const float* W_x    = (const float*)d_in[4];  // (96, DI)
    const float* W_dt   = (const float*)d_in[5];  // (DI, DTR)
    const float* b_dt   = (const float*)d_in[6];  // (DI)
    const float* A_log  = (const float*)d_in[7];  // (DI, DS)
    const float* Dvec   = (const float*)d_in[8];  // (DI)
    const float* W_out  = (const float*)d_in[9];  // (DM, DI)
    float* out = (float*)d_out;

    // Workspace layout (floats): ~136 MB total
    float* ws    = (float*)d_ws;
    float* xz    = ws;                                   // (B,T,2*DI) = 16.8M
    float* x_act = xz    + (size_t)MTOT * 2 * DI;        // (B,T,DI)   =  8.4M
    float* xdbl  = x_act + (size_t)MTOT * DI;            // (B,T,96)   =  0.4M
    float* dtb   = xdbl  + (size_t)MTOT * XDBL_W;        // (B,T,DI)   =  8.4M

    dim3 blk(256);

    // 1) in_proj: xz = hs @ W_in^T   (M=4096, N=4096, K=1024)
    gemm_nt_wmma<0><<<dim3((2 * DI) / 256, MTOT / 64), blk, 0, stream>>>(
        hs, DM, W_in, DM, xz, 2 * DI, DM, nullptr);

    // 2) depthwise causal conv + SiLU -> x_act
    conv_silu_kernel<<<dim3((MTOT * DI) / 256), blk, 0, stream>>>(
        xz, conv_w, conv_b, x_act);

    // 3) x_dbl = x_act @ W_x^T       (M=4096, N=96, K=2048)  [ragged N]
    gemm_nt_wmma_ragged<<<dim3(1, MTOT / 32), blk, 0, stream>>>(
        x_act, DI, W_x, DI, xdbl, XDBL_W, XDBL_W, DI);

    // 4) dt = softplus(x_dbl[:, :DTR] @ W_dt^T + b_dt)  (M=4096, N=2048, K=64)
    gemm_nt_wmma<1><<<dim3(DI / 256, MTOT / 64), blk, 0, stream>>>(
        xdbl, XDBL_W, W_dt, DTR, dtb, DI, DTR, b_dt);

    // 5) selective scan + skip + gating (writes gated y in place into x_act)
    scan_kernel<<<dim3((BSZ * DI * DS) / 256), blk, 0, stream>>>(
        dtb, xdbl, xz, A_log, Dvec, x_act);

    // 6) out = y @ W_out^T           (M=4096, N=1024, K=2048)
    gemm_nt_wmma<0><<<dim3(DM / 256, MTOT / 64), blk, 0, stream>>>(
        x_act, DI, W_out, DI, out, DM, DI, nullptr);
}